// LFVULoss_4518305595626
// MI455X (gfx1250) — compile-verified
//
#include <hip/hip_runtime.h>
#include <math.h>

// Problem constants (from reference setup_inputs)
#define NPTS   8192
#define DIN    512
#define DLAT   32
#define KNN    10
#define EPSF   1e-8f
#define MROWS  32     // rows per block (2 WMMA M-tiles)

// LDS strides chosen so (stride mod 64) == 4 -> the 16 row-lanes of a WMMA
// fragment b64 read hit distinct bank groups (no conflicts).
#define SA 516   // A tile stride (full K = 512)
#define SB 260   // B tile stride (K chunk = 256)

typedef __attribute__((ext_vector_type(2))) float v2f;
typedef __attribute__((ext_vector_type(8))) float v8f;

#define WMMA_F32(a, b, c) \
  __builtin_amdgcn_wmma_f32_16x16x4_f32(false, (a), false, (b), (short)0, (c), false, false)

// ---------------------------------------------------------------------------
// Kernel 0: row norms ||x_i||^2, one wave32 per row.
// ---------------------------------------------------------------------------
__global__ __launch_bounds__(256) void lfvu_norms(const float* __restrict__ X,
                                                  float* __restrict__ norms) {
  const int lane = threadIdx.x & 31;
  const int wave = threadIdx.x >> 5;
  const int row  = blockIdx.x * 8 + wave;
  const float* xr = X + (size_t)row * DIN;
  float s = 0.f;
#pragma unroll
  for (int i = 0; i < DIN / 32; ++i) {
    float v = xr[lane + 32 * i];
    s = fmaf(v, v, s);
  }
#pragma unroll
  for (int off = 16; off > 0; off >>= 1) s += __shfl_xor(s, off, 32);
  if (lane == 0) norms[row] = s;
}

// ---------------------------------------------------------------------------
// Kernel 1: fused  G = X·Xt (WMMA f32)  ->  d = sqrt(clip(na+nb-2g)+eps)
//           -> per-row top-10 selection -> latent dists -> LFVU per row.
// Block = 32 rows (two M-tiles), 4 waves; each wave owns one 16-col subtile
// of the 64-col B tile and both M-tiles (4 independent WMMA chains).
// B tiles staged via CDNA5 async global->LDS loads (ASYNCcnt).
// ---------------------------------------------------------------------------
__global__ __launch_bounds__(128) void lfvu_main(const float* __restrict__ Xo,
                                                 const float* __restrict__ Xl,
                                                 const float* __restrict__ norms,
                                                 float* __restrict__ partials) {
  __shared__ float ldsA[MROWS * SA];       // 66.0 KB  A rows, full K
  __shared__ float ldsB[64 * SB];          // 66.5 KB  B rows, K chunk of 256
  __shared__ float scr[4][MROWS * 17];     //  8.7 KB  per-wave tile transpose
  __shared__ float mergeD[MROWS][4 * KNN]; //  5.0 KB  4 lists x 10 per row
  __shared__ int   mergeI[MROWS][4 * KNN]; //  5.0 KB
  __shared__ float lat[MROWS][DLAT];       //  4.0 KB  own latent rows
  __shared__ float rowRes[MROWS];

  const int tid  = threadIdx.x;
  const int lane = tid & 31;
  const int wave = tid >> 5;
  const int row0 = blockIdx.x * MROWS;

  const int hh = lane >> 4;   // lane half: selects K sub-pair / M offset 8
  const int nn = lane & 15;   // N (or M) index within 16

  // ---- stage A (32 x 512) into LDS, coalesced float4 ----
  for (int g = tid; g < MROWS * (DIN / 4); g += 128) {
    const int r = g >> 7, kq = g & 127;
    const float4 v = ((const float4*)(Xo + (size_t)(row0 + r) * DIN))[kq];
    *(float4*)&ldsA[r * SA + kq * 4] = v;
  }
  // ---- stage this block's latent rows ----
  for (int g = tid; g < MROWS * DLAT; g += 128)
    lat[g >> 5][g & 31] = Xl[(size_t)(row0 + (g >> 5)) * DLAT + (g & 31)];

  // ---- per-lane row norms for the two M-tiles ----
  float na0[8], na1[8];
#pragma unroll
  for (int v = 0; v < 8; ++v) {
    na0[v] = norms[row0 + v + 8 * hh];
    na1[v] = norms[row0 + 16 + v + 8 * hh];
  }

  // ---- per-lane running top-10 (sorted ascending, register resident) ----
  float bestd[KNN]; int besti[KNN];
#pragma unroll
  for (int j = 0; j < KNN; ++j) { bestd[j] = 3.4e38f; besti[j] = -1; }

  const int rL = lane;        // row (0..31) this lane owns after transpose

  __syncthreads();

  for (int nt = 0; nt < NPTS / 64; ++nt) {
    const int cb = nt * 64;
    v8f c00 = {}, c01 = {}, c10 = {}, c11 = {};

    for (int kc = 0; kc < DIN; kc += 256) {
      __syncthreads();   // previous compute done before overwriting ldsB
      // ---- async stage B chunk: 64 rows x 256 K via global->LDS DMA ----
      {
        const float* gbase = Xo + (size_t)cb * DIN + kc;   // uniform SGPR base
        for (int g = tid; g < 64 * 64; g += 128) {
          const int r = g >> 6, kq = g & 63;
          const unsigned voff   = (unsigned)(r * (DIN * 4) + kq * 16);
          const unsigned ldsoff = (unsigned)(size_t)&ldsB[r * SB + kq * 4];
          asm volatile("global_load_async_to_lds_b128 %0, %1, %2"
                       :: "v"(ldsoff), "v"(voff), "s"(gbase) : "memory");
        }
        asm volatile("s_wait_asynccnt 0x0" ::: "memory");
      }
      __syncthreads();

      const float* pA0 = &ldsA[nn * SA + kc + 2 * hh];
      const float* pA1 = &ldsA[(16 + nn) * SA + kc + 2 * hh];
      const float* pB  = &ldsB[(wave * 16 + nn) * SB + 2 * hh];
#pragma unroll 8
      for (int k = 0; k < 256; k += 8) {
        v2f b0  = *(const v2f*)(pB + k);
        v2f b1  = *(const v2f*)(pB + k + 4);
        v2f a00 = *(const v2f*)(pA0 + k);
        v2f a10 = *(const v2f*)(pA1 + k);
        v2f a01 = *(const v2f*)(pA0 + k + 4);
        v2f a11 = *(const v2f*)(pA1 + k + 4);
        c00 = WMMA_F32(a00, b0, c00);
        c10 = WMMA_F32(a10, b0, c10);
        c01 = WMMA_F32(a01, b1, c01);
        c11 = WMMA_F32(a11, b1, c11);
      }
    }
    const v8f cM0 = c00 + c01;
    const v8f cM1 = c10 + c11;

    // ---- epilogue: distances for this wave's 32x16 tile ----
    const int colBase = cb + wave * 16;
    const float nbv = norms[colBase + nn];   // norm of this lane's column
    float* s = &scr[wave][0];
#pragma unroll
    for (int v = 0; v < 8; ++v) {
      float d2a = na0[v] + nbv - 2.0f * cM0[v];
      float d2b = na1[v] + nbv - 2.0f * cM1[v];
      s[(v + 8 * hh) * 17 + nn]      = sqrtf(fmaxf(d2a, 0.0f) + EPSF);
      s[(16 + v + 8 * hh) * 17 + nn] = sqrtf(fmaxf(d2b, 0.0f) + EPSF);
    }
    // wave-synchronous transpose read: lane rL owns row rL, all 16 cols
#pragma unroll
    for (int c = 0; c < 16; ++c) {
      const int gcol = colBase + c;
      const float d  = s[rL * 17 + c];
      if (gcol == row0 + rL) continue;           // exclude self
      if (d < bestd[KNN - 1]) {
        float cd = d; int ci = gcol;
#pragma unroll
        for (int q = 0; q < KNN; ++q) {
          if (cd < bestd[q]) {
            float td = bestd[q]; bestd[q] = cd; cd = td;
            int   ti = besti[q]; besti[q] = ci; ci = ti;
          }
        }
      }
    }
  }

  // ---- dump 4 partial lists per row (one per wave) ----
#pragma unroll
  for (int j = 0; j < KNN; ++j) {
    mergeD[rL][wave * KNN + j] = bestd[j];
    mergeI[rL][wave * KNN + j] = besti[j];
  }
  __syncthreads();

  // ---- per-row merge + latent dists + LFVU (one thread per row) ----
  if (tid < MROWS) {
    const int r = tid;
    float kd[KNN]; int ki[KNN];
#pragma unroll
    for (int j = 0; j < KNN; ++j) { kd[j] = 3.4e38f; ki[j] = -1; }
    for (int p = 0; p < 4 * KNN; ++p) {
      float d = mergeD[r][p];
      if (d < kd[KNN - 1]) {
        float cd = d; int ci = mergeI[r][p];
#pragma unroll
        for (int q = 0; q < KNN; ++q) {
          if (cd < kd[q]) {
            float td = kd[q]; kd[q] = cd; cd = td;
            int   ti = ki[q]; ki[q] = ci; ci = ti;
          }
        }
      }
    }
    // latent distances to the 10 neighbors
    float rs[KNN];
#pragma unroll
    for (int j = 0; j < KNN; ++j) {
      const float* xl = Xl + (size_t)ki[j] * DLAT;
      float acc = 0.f;
#pragma unroll
      for (int cix = 0; cix < DLAT; ++cix) {
        float df = lat[r][cix] - xl[cix];
        acc = fmaf(df, df, acc);
      }
      rs[j] = sqrtf(acc);
    }
    // normalize (mean / std ddof=1 + eps) and LFVU
    float md = 0.f, mr = 0.f;
#pragma unroll
    for (int j = 0; j < KNN; ++j) { md += kd[j]; mr += rs[j]; }
    md *= (1.f / KNN); mr *= (1.f / KNN);
    float vd = 0.f, vr = 0.f;
#pragma unroll
    for (int j = 0; j < KNN; ++j) {
      float a = kd[j] - md; vd = fmaf(a, a, vd);
      float b = rs[j] - mr; vr = fmaf(b, b, vr);
    }
    const float sd = sqrtf(vd / (KNN - 1)) + EPSF;
    const float sr = sqrtf(vr / (KNN - 1)) + EPSF;
    float dn[KNN], num = 0.f, mdn = 0.f;
#pragma unroll
    for (int j = 0; j < KNN; ++j) {
      dn[j] = (kd[j] - md) / sd;
      float rn = (rs[j] - mr) / sr;
      float e = dn[j] - rn;
      num = fmaf(e, e, num);
      mdn += dn[j];
    }
    mdn *= (1.f / KNN);
    float den = 0.f;
#pragma unroll
    for (int j = 0; j < KNN; ++j) {
      float cdev = dn[j] - mdn;
      den = fmaf(cdev, cdev, den);
    }
    den = fmaxf(den, EPSF);
    rowRes[r] = num / den;
  }
  __syncthreads();
  if (tid == 0) {
    float s = 0.f;
#pragma unroll
    for (int r = 0; r < MROWS; ++r) s += rowRes[r];
    partials[blockIdx.x] = s;
  }
}

// ---------------------------------------------------------------------------
// Kernel 2: deterministic final reduction of 256 block partials.
// ---------------------------------------------------------------------------
__global__ __launch_bounds__(256) void lfvu_finalize(const float* __restrict__ partials,
                                                     float* __restrict__ out) {
  __shared__ float buf[NPTS / MROWS];
  const int t = threadIdx.x;
  buf[t] = partials[t];
  __syncthreads();
  for (int s = (NPTS / MROWS) / 2; s > 0; s >>= 1) {
    if (t < s) buf[t] += buf[t + s];
    __syncthreads();
  }
  if (t == 0) out[0] = buf[0] * (1.0f / NPTS);   // ALPHA = 1
}

// ---------------------------------------------------------------------------
extern "C" void kernel_launch(void* const* d_in, const int* in_sizes, int n_in,
                              void* d_out, int out_size, void* d_ws, size_t ws_size,
                              hipStream_t stream) {
  (void)in_sizes; (void)n_in; (void)out_size; (void)ws_size;
  const float* Xo = (const float*)d_in[0];   // 8192 x 512 fp32
  const float* Xl = (const float*)d_in[1];   // 8192 x 32  fp32
  float* out      = (float*)d_out;

  float* norms    = (float*)d_ws;            // 8192 floats
  float* partials = norms + NPTS;            // 256 floats

  lfvu_norms<<<NPTS / 8, 256, 0, stream>>>(Xo, norms);
  lfvu_main<<<NPTS / MROWS, 128, 0, stream>>>(Xo, Xl, norms, partials);
  lfvu_finalize<<<1, NPTS / MROWS, 0, stream>>>(partials, out);
}